// GNNStack_63350767616471
// MI455X (gfx1250) — compile-verified
//
#include <hip/hip_runtime.h>
#include <hip/hip_bf16.h>

#define HEADS 5
#define CPH   32
#define FDIM  160   // HEADS*CPH

typedef __attribute__((ext_vector_type(2))) float v2f;
typedef __attribute__((ext_vector_type(8))) float v8f;

// ------------------------------------------------------------------
// WMMA f32 GEMM:  H[N,M] = X[N,K] @ W[K,M]; N%16==0, M%16==0, K%4==0
// One wave per 16x16 output tile, V_WMMA_F32_16X16X4_F32 along K.
// A 16x4 layout: lane&15 = row M, lane>>4 selects K pair, 2 VGPRs (v2f).
// B 4x16 layout mirrored. C/D: 8 VGPRs, vgpr j -> M = j + 8*(lane>>4).
// ------------------------------------------------------------------
__global__ void gemm_xw_wmma(const float* __restrict__ X, const float* __restrict__ W,
                             float* __restrict__ Hout, int N, int K, int M) {
  const int tilesM = M >> 4;
  const int totalTiles = (N >> 4) * tilesM;
  int wave = (int)(((long long)blockIdx.x * blockDim.x + threadIdx.x) >> 5);
  if (wave >= totalTiles) return;          // wave-uniform: EXEC stays all-1s inside
  int lane = threadIdx.x & 31;
  int tm = wave % tilesM;
  int tn = wave / tilesM;
  int r  = lane & 15;
  int kh = (lane >> 4) << 1;               // 0 or 2
  const float* Xrow = X + (size_t)(tn * 16 + r) * K;
  const float* Wcol = W + (tm * 16 + r);
  v8f acc = {};
  for (int k = 0; k < K; k += 4) {
    v2f a, b;
    a.x = Xrow[k + kh];
    a.y = Xrow[k + kh + 1];
    b.x = Wcol[(size_t)(k + kh) * M];
    b.y = Wcol[(size_t)(k + kh + 1) * M];
    acc = __builtin_amdgcn_wmma_f32_16x16x4_f32(false, a, false, b, (short)0, acc,
                                                false, false);
  }
  int orow = tn * 16 + ((lane >> 4) << 3);
  float* O = Hout + (size_t)orow * M + (tm * 16 + r);
#pragma unroll
  for (int j = 0; j < 8; ++j) O[(size_t)j * M] = acc[j];
}

// ---------------- per-node attention logits: al[n,h] = sum_c h[n,h*32+c]*a[h*32+c]
__global__ void node_logits(const float* __restrict__ Hf, const float* __restrict__ avec,
                            float* __restrict__ out, int N) {
  long long i = (long long)blockIdx.x * blockDim.x + threadIdx.x;
  if (i >= (long long)N * HEADS) return;
  int n = (int)(i / HEADS), h = (int)(i % HEADS);
  const float* hp = Hf + (size_t)n * FDIM + h * CPH;
  const float* ap = avec + h * CPH;
  float s = 0.f;
#pragma unroll
  for (int c = 0; c < CPH; ++c) s += hp[c] * ap[c];
  out[i] = s;
}

// ---------------- wedot[h] = sum_c We[h*32+c]*a_edge[h*32+c]
__global__ void compute_wedot(const float* __restrict__ We, const float* __restrict__ ae,
                              float* __restrict__ wd) {
  int h = threadIdx.x;
  if (h < HEADS) {
    float s = 0.f;
    for (int c = 0; c < CPH; ++c) s += We[h * CPH + c] * ae[h * CPH + c];
    wd[h] = s;
  }
}

__global__ void fill_f32(float* __restrict__ p, float v, long long n) {
  long long i = (long long)blockIdx.x * blockDim.x + threadIdx.x;
  if (i < n) p[i] = v;
}

__device__ __forceinline__ void atomicMaxFloat(float* addr, float val) {
  if (val >= 0.f) atomicMax((int*)addr, __float_as_int(val));
  else            atomicMin((unsigned int*)addr, (unsigned int)__float_as_int(val));
}

// raw = leakyrelu(alS[row]+alD[col]+ew*wedot, 0.2); store raw; segment max into m
__global__ void edge_raw_max(const float* __restrict__ alS, const float* __restrict__ alD,
                             const float* __restrict__ ew, const float* __restrict__ wd,
                             const long long* __restrict__ gat, const long long* __restrict__ seg,
                             float* __restrict__ eraw, float* __restrict__ m, long long E) {
  long long i = (long long)blockIdx.x * blockDim.x + threadIdx.x;
  if (i >= E * HEADS) return;
  long long e = i / HEADS; int h = (int)(i % HEADS);
  long long s = gat[e], d = seg[e];
  float raw = alS[s * HEADS + h] + alD[d * HEADS + h] + ew[e] * wd[h];
  raw = raw > 0.f ? raw : 0.2f * raw;
  eraw[i] = raw;
  atomicMaxFloat(&m[d * HEADS + h], raw);
}

// m = isfinite(m) ? m : 0   (nodes with no in-edges stay at -inf)
__global__ void fix_m(float* __restrict__ m, long long n) {
  long long i = (long long)blockIdx.x * blockDim.x + threadIdx.x;
  if (i >= n) return;
  float v = m[i];
  if (!(v >= -3.0e38f && v <= 3.0e38f)) m[i] = 0.f;
}

// e = exp(raw - m[seg]); store e; segment-sum into denom
__global__ void edge_exp(float* __restrict__ eex, const float* __restrict__ m,
                         const long long* __restrict__ seg, float* __restrict__ denom,
                         long long E) {
  long long i = (long long)blockIdx.x * blockDim.x + threadIdx.x;
  if (i >= E * HEADS) return;
  long long e = i / HEADS; int h = (int)(i % HEADS);
  float ex = expf(eex[i] - m[seg[e] * HEADS + h]);
  eex[i] = ex;
  atomicAdd(&denom[seg[e] * HEADS + h], ex);
}

__global__ void write_alpha(const float* __restrict__ eex, const float* __restrict__ denom,
                            const long long* __restrict__ seg, float* __restrict__ out,
                            long long E) {
  long long i = (long long)blockIdx.x * blockDim.x + threadIdx.x;
  if (i >= E * HEADS) return;
  long long e = i / HEADS; int h = (int)(i % HEADS);
  out[i] = eex[i] / (denom[seg[e] * HEADS + h] + 1e-16f);
}

// agg[col, hc] += hs[row, hc] * alpha[e,h]   (thread per (edge, channel))
__global__ void edge_agg(const float* __restrict__ hs, const float* __restrict__ eex,
                         const float* __restrict__ denom,
                         const long long* __restrict__ gat, const long long* __restrict__ seg,
                         float* __restrict__ agg, long long E) {
  long long i = (long long)blockIdx.x * blockDim.x + threadIdx.x;
  if (i >= E * FDIM) return;
  long long e = i / FDIM;
  int hc = (int)(i % FDIM);
  int h = hc >> 5;
  long long r = gat[e], d = seg[e];
  float a = eex[e * HEADS + h] / (denom[d * HEADS + h] + 1e-16f);
  atomicAdd(&agg[d * FDIM + hc], hs[r * FDIM + hc] * a);
}

// concat epilogue: x = LN(leakyrelu(agg + b, 0.01)) over 160; wave per node
__global__ void epilogue_concat_ln(const float* __restrict__ agg, const float* __restrict__ bias,
                                   const float* __restrict__ g, const float* __restrict__ bt,
                                   float* __restrict__ out, int N) {
  int node = (int)(((long long)blockIdx.x * blockDim.x + threadIdx.x) >> 5);
  int lane = threadIdx.x & 31;
  if (node >= N) return;
  const float* p = agg + (size_t)node * FDIM;
  float v[5]; float s = 0.f;
#pragma unroll
  for (int j = 0; j < 5; ++j) {
    int c = lane + 32 * j;
    float x = p[c] + bias[c];
    x = x > 0.f ? x : 0.01f * x;
    v[j] = x; s += x;
  }
#pragma unroll
  for (int off = 16; off > 0; off >>= 1) s += __shfl_xor(s, off, 32);
  float mu = s * (1.f / FDIM);
  float var = 0.f;
#pragma unroll
  for (int j = 0; j < 5; ++j) { float d = v[j] - mu; var += d * d; }
#pragma unroll
  for (int off = 16; off > 0; off >>= 1) var += __shfl_xor(var, off, 32);
  float inv = rsqrtf(var * (1.f / FDIM) + 1e-5f);
  float* o = out + (size_t)node * FDIM;
#pragma unroll
  for (int j = 0; j < 5; ++j) {
    int c = lane + 32 * j;
    o[c] = (v[j] - mu) * inv * g[c] + bt[c];
  }
}

// mean-over-heads epilogue: 32 channels/node, lane = channel
__global__ void epilogue_mean_ln(const float* __restrict__ agg, const float* __restrict__ bias,
                                 const float* __restrict__ g, const float* __restrict__ bt,
                                 float* __restrict__ out, int N) {
  int node = (int)(((long long)blockIdx.x * blockDim.x + threadIdx.x) >> 5);
  int c = threadIdx.x & 31;
  if (node >= N) return;
  const float* p = agg + (size_t)node * FDIM;
  float x = 0.f;
#pragma unroll
  for (int h = 0; h < HEADS; ++h) x += p[h * 32 + c];
  x = x * 0.2f + bias[c];                  // mean over 5 heads
  x = x > 0.f ? x : 0.01f * x;
  float s = x;
#pragma unroll
  for (int off = 16; off > 0; off >>= 1) s += __shfl_xor(s, off, 32);
  float mu = s * (1.f / 32.f);
  float d = x - mu;
  float var = d * d;
#pragma unroll
  for (int off = 16; off > 0; off >>= 1) var += __shfl_xor(var, off, 32);
  out[(size_t)node * 32 + c] = d * rsqrtf(var * (1.f / 32.f) + 1e-5f) * g[c] + bt[c];
}

// ------------------------------------------------------------------
static inline unsigned cdivu(long long a, long long b) { return (unsigned)((a + b - 1) / b); }

static void run_conv(hipStream_t st,
                     const float* hSrc, const float* hDst,
                     const float* aS, const float* aD, const float* aE, const float* We,
                     const float* ew, const long long* gat, const long long* seg,
                     long long E, int nSrc, int nDst,
                     float* alA, float* alB, float* mbuf, float* denom,
                     float* eex, float* agg, float* wedot, float* alphaOut) {
  long long EH = E * HEADS;
  node_logits<<<cdivu((long long)nSrc * HEADS, 256), 256, 0, st>>>(hSrc, aS, alA, nSrc);
  node_logits<<<cdivu((long long)nDst * HEADS, 256), 256, 0, st>>>(hDst, aD, alB, nDst);
  compute_wedot<<<1, 32, 0, st>>>(We, aE, wedot);
  fill_f32<<<cdivu((long long)nDst * HEADS, 256), 256, 0, st>>>(mbuf, -__builtin_inff(), (long long)nDst * HEADS);
  fill_f32<<<cdivu((long long)nDst * HEADS, 256), 256, 0, st>>>(denom, 0.f, (long long)nDst * HEADS);
  fill_f32<<<cdivu((long long)nDst * FDIM, 256), 256, 0, st>>>(agg, 0.f, (long long)nDst * FDIM);
  edge_raw_max<<<cdivu(EH, 256), 256, 0, st>>>(alA, alB, ew, wedot, gat, seg, eex, mbuf, E);
  fix_m<<<cdivu((long long)nDst * HEADS, 256), 256, 0, st>>>(mbuf, (long long)nDst * HEADS);
  edge_exp<<<cdivu(EH, 256), 256, 0, st>>>(eex, mbuf, seg, denom, E);
  if (alphaOut)
    write_alpha<<<cdivu(EH, 256), 256, 0, st>>>(eex, denom, seg, alphaOut, E);
  edge_agg<<<cdivu(E * FDIM, 256), 256, 0, st>>>(hSrc, eex, denom, gat, seg, agg, E);
}

extern "C" void kernel_launch(void* const* d_in, const int* in_sizes, int n_in,
                              void* d_out, int out_size, void* d_ws, size_t ws_size,
                              hipStream_t stream) {
  const float*      x_s  = (const float*)d_in[0];
  const float*      x_t  = (const float*)d_in[1];
  const long long*  ei   = (const long long*)d_in[2];
  const float*      ew   = (const float*)d_in[3];
  const float*      W1   = (const float*)d_in[4];
  const float*      a1s  = (const float*)d_in[5];
  const float*      a1d  = (const float*)d_in[6];
  const float*      a1e  = (const float*)d_in[7];
  const float*      We1  = (const float*)d_in[8];
  const float*      b1   = (const float*)d_in[9];
  const float*      g0   = (const float*)d_in[10];
  const float*      bt0  = (const float*)d_in[11];
  const float*      W2s  = (const float*)d_in[12];
  const float*      W2d  = (const float*)d_in[13];
  const float*      a2s  = (const float*)d_in[14];
  const float*      a2d  = (const float*)d_in[15];
  const float*      a2e  = (const float*)d_in[16];
  const float*      We2  = (const float*)d_in[17];
  const float*      b2   = (const float*)d_in[18];
  const float*      g1   = (const float*)d_in[19];
  const float*      bt1  = (const float*)d_in[20];

  int       n_s = in_sizes[0] / 64;
  int       n_t = in_sizes[1] / 64;
  long long E   = in_sizes[3];
  const long long* row = ei;
  const long long* col = ei + E;
  int nmax = n_s > n_t ? n_s : n_t;

  // workspace carve-up (floats)
  float* w = (float*)d_ws;
  size_t off = 0;
  auto alloc = [&](size_t nf) { float* p = w + off; off += nf; return p; };
  float* hA    = alloc((size_t)nmax * FDIM);
  float* hB    = alloc((size_t)nmax * FDIM);
  float* x1t   = alloc((size_t)n_t * FDIM);
  float* x1s   = alloc((size_t)n_s * FDIM);
  float* agg   = alloc((size_t)nmax * FDIM);
  float* eex   = alloc((size_t)E * HEADS);
  float* alA   = alloc((size_t)nmax * HEADS);
  float* alB   = alloc((size_t)nmax * HEADS);
  float* mbuf  = alloc((size_t)nmax * HEADS);
  float* denom = alloc((size_t)nmax * HEADS);
  float* wedot = alloc(8);

  // output carve-up: (x_s, at_s, x_t, at_t)
  float* out_xs  = (float*)d_out;
  float* out_ats = out_xs + (size_t)n_s * 32;
  float* out_xt  = out_ats + (size_t)E * HEADS;
  float* out_att = out_xt + (size_t)n_t * 32;

  auto gemm = [&](const float* X, const float* Wm, float* Hout, int N, int K) {
    int tiles = (N >> 4) * (FDIM >> 4);
    gemm_xw_wmma<<<cdivu(tiles, 8), 256, 0, stream>>>(X, Wm, Hout, N, K, FDIM);
  };

  // ---------------- layer 1 (shared W1 for src/dst) ----------------
  gemm(x_s, W1, hA, n_s, 64);
  gemm(x_t, W1, hB, n_t, 64);
  // conv -> t side: src=x_s(hA), dst=x_t(hB), gather=row, segment=col
  run_conv(stream, hA, hB, a1s, a1d, a1e, We1, ew, row, col, E, n_s, n_t,
           alA, alB, mbuf, denom, eex, agg, wedot, nullptr);
  epilogue_concat_ln<<<cdivu((long long)n_t * 32, 256), 256, 0, stream>>>(agg, b1, g0, bt0, x1t, n_t);
  // conv -> s side: src=x_t(hB), dst=x_s(hA), gather=col, segment=row
  run_conv(stream, hB, hA, a1s, a1d, a1e, We1, ew, col, row, E, n_t, n_s,
           alA, alB, mbuf, denom, eex, agg, wedot, nullptr);
  epilogue_concat_ln<<<cdivu((long long)n_s * 32, 256), 256, 0, stream>>>(agg, b1, g0, bt0, x1s, n_s);

  // ---------------- layer 2 (separate W2s/W2d, concat=False) -------
  // conv -> t side: hs = x1s@W2s, hd = x1t@W2d
  gemm(x1s, W2s, hA, n_s, FDIM);
  gemm(x1t, W2d, hB, n_t, FDIM);
  run_conv(stream, hA, hB, a2s, a2d, a2e, We2, ew, row, col, E, n_s, n_t,
           alA, alB, mbuf, denom, eex, agg, wedot, out_att);
  epilogue_mean_ln<<<cdivu((long long)n_t * 32, 256), 256, 0, stream>>>(agg, b2, g1, bt1, out_xt, n_t);
  // conv -> s side: hs = x1t@W2s, hd = x1s@W2d
  gemm(x1t, W2s, hA, n_t, FDIM);
  gemm(x1s, W2d, hB, n_s, FDIM);
  run_conv(stream, hA, hB, a2s, a2d, a2e, We2, ew, col, row, E, n_t, n_s,
           alA, alB, mbuf, denom, eex, agg, wedot, out_ats);
  epilogue_mean_ln<<<cdivu((long long)n_s * 32, 256), 256, 0, stream>>>(agg, b2, g1, bt1, out_xs, n_s);
}